// GINEConvLayer_13048110645792
// MI455X (gfx1250) — compile-verified
//
#include <hip/hip_runtime.h>

#define N_NODES 50000
#define N_EDGES 800000
#define DIM 128
#define BN_EPS 1e-5f

typedef __attribute__((ext_vector_type(2))) float v2f;
typedef __attribute__((ext_vector_type(8))) float v8f;

// ---------------------------------------------------------------------------
// Kernel 1: msg = relu(x[src] + edge_attr), atomically accumulated into h
// (h was pre-initialized to x, so afterwards h = x + aggr, GIN eps = 0).
// 32 lanes per edge, 4 floats per lane.
// ---------------------------------------------------------------------------
__global__ __launch_bounds__(256)
void edge_scatter(const float* __restrict__ x,
                  const float* __restrict__ ea,
                  const long long* __restrict__ ei,
                  float* __restrict__ h) {
  int t = blockIdx.x * blockDim.x + threadIdx.x;
  int e = t >> 5;
  int q = (t & 31) << 2;           // feature offset 0..124
  long long src = ei[e];
  long long dst = ei[N_EDGES + e];
  float4 xa = *(const float4*)(x + (size_t)src * DIM + q);
  float4 eb = *(const float4*)(ea + (size_t)e * DIM + q);
  float4 m;
  m.x = fmaxf(xa.x + eb.x, 0.0f);
  m.y = fmaxf(xa.y + eb.y, 0.0f);
  m.z = fmaxf(xa.z + eb.z, 0.0f);
  m.w = fmaxf(xa.w + eb.w, 0.0f);
  float* hp = h + (size_t)dst * DIM + q;
  atomicAdd(hp + 0, m.x);
  atomicAdd(hp + 1, m.y);
  atomicAdd(hp + 2, m.z);
  atomicAdd(hp + 3, m.w);
}

// ---------------------------------------------------------------------------
// GEMM1: h1 = h @ W1^T + b1.  One block = 16 rows x 128 cols; 8 waves each
// own a 16x16 output tile and run 32 x V_WMMA_F32_16X16X4_F32 over K=128.
// A fragment layout (ISA 7.12.2, 32-bit A 16x4): lane L<16 holds M=L,
// {K=k, K=k+1} in {v0,v1}; lane L>=16 holds M=L-16, {K=k+2, K=k+3}.
// B fragment (4x16): lane L<16 holds {K=k,K=k+1} x N=L; lanes>=16 K=k+2/k+3.
// D layout: VGPR r -> row r (lanes 0-15) / row r+8 (lanes 16-31), col = lane&15.
// ---------------------------------------------------------------------------
__global__ __launch_bounds__(256)
void gemm1_wmma(const float* __restrict__ h,
                const float* __restrict__ W1,
                const float* __restrict__ b1,
                float* __restrict__ h1) {
  __shared__ float Ah[16][132];                 // stride 132 -> conflict-free
  int row0 = blockIdx.x * 16;
  int tid = threadIdx.x;
  {                                             // cooperative 16x128 load
    int r = tid >> 4;
    int c = (tid & 15) * 8;
    const float4* src = (const float4*)(h + (size_t)(row0 + r) * DIM + c);
    float4 v0 = src[0], v1 = src[1];
    Ah[r][c + 0] = v0.x; Ah[r][c + 1] = v0.y; Ah[r][c + 2] = v0.z; Ah[r][c + 3] = v0.w;
    Ah[r][c + 4] = v1.x; Ah[r][c + 5] = v1.y; Ah[r][c + 6] = v1.z; Ah[r][c + 7] = v1.w;
  }
  __syncthreads();

  int wave = tid >> 5;
  int lane = tid & 31;
  int n    = lane & 15;             // also the A-row index m
  int kh   = (lane >> 4) << 1;      // 0 or 2
  int col0 = wave * 16;
  const float* Wrow = W1 + (size_t)(col0 + n) * DIM;  // W1[o][i], o = col0+n

  v8f acc = {};
  #pragma unroll
  for (int k = 0; k < DIM; k += 4) {
    v2f a, b;
    a.x = Ah[n][k + kh];
    a.y = Ah[n][k + kh + 1];
    b.x = Wrow[k + kh];
    b.y = Wrow[k + kh + 1];
    acc = __builtin_amdgcn_wmma_f32_16x16x4_f32(false, a, false, b,
                                                (short)0, acc, false, false);
  }

  int half = lane >> 4;
  float bias = b1[col0 + n];
  #pragma unroll
  for (int r = 0; r < 8; ++r) {
    int row = row0 + r + 8 * half;
    h1[(size_t)row * DIM + col0 + n] = acc[r] + bias;
  }
}

// ---------------------------------------------------------------------------
// Per-column sum / sum-of-squares partial reduction (128 threads = 1 per col,
// 250 rows per block, 200 blocks -> exactly 50000 rows).
// ---------------------------------------------------------------------------
__global__ __launch_bounds__(128)
void colstats(const float* __restrict__ h1,
              float* __restrict__ csum, float* __restrict__ csq) {
  int c = threadIdx.x;
  int rbase = blockIdx.x * 250;
  float s = 0.0f, ss = 0.0f;
  for (int i = 0; i < 250; ++i) {
    float v = h1[(size_t)(rbase + i) * DIM + c];
    s += v;
    ss += v * v;
  }
  atomicAdd(&csum[c], s);
  atomicAdd(&csq[c], ss);
}

// Fold BN into per-feature scale/shift: y = s*x + t
__global__ __launch_bounds__(128)
void bnparams(const float* __restrict__ csum, const float* __restrict__ csq,
              const float* __restrict__ gamma, const float* __restrict__ beta,
              float* __restrict__ sarr, float* __restrict__ tarr) {
  int c = threadIdx.x;
  float invn = 1.0f / (float)N_NODES;
  float mean = csum[c] * invn;
  float var  = csq[c] * invn - mean * mean;
  float sc   = gamma[c] * rsqrtf(var + BN_EPS);
  sarr[c] = sc;
  tarr[c] = beta[c] - mean * sc;
}

// ---------------------------------------------------------------------------
// GEMM2: out = x + relu( relu(BN(h1)) @ W2^T + b2 ).  BN+ReLU applied while
// staging the A tile into LDS; same WMMA structure as GEMM1.
// ---------------------------------------------------------------------------
__global__ __launch_bounds__(256)
void gemm2_wmma(const float* __restrict__ h1,
                const float* __restrict__ sarr, const float* __restrict__ tarr,
                const float* __restrict__ W2,
                const float* __restrict__ b2,
                const float* __restrict__ x,
                float* __restrict__ out) {
  __shared__ float Ah[16][132];
  int row0 = blockIdx.x * 16;
  int tid = threadIdx.x;
  {
    int r = tid >> 4;
    int c = (tid & 15) * 8;
    const float4* src = (const float4*)(h1 + (size_t)(row0 + r) * DIM + c);
    float4 v0 = src[0], v1 = src[1];
    float tmp[8] = {v0.x, v0.y, v0.z, v0.w, v1.x, v1.y, v1.z, v1.w};
    #pragma unroll
    for (int j = 0; j < 8; ++j) {
      float v = fmaxf(tmp[j] * sarr[c + j] + tarr[c + j], 0.0f);
      Ah[r][c + j] = v;
    }
  }
  __syncthreads();

  int wave = tid >> 5;
  int lane = tid & 31;
  int n    = lane & 15;
  int kh   = (lane >> 4) << 1;
  int col0 = wave * 16;
  const float* Wrow = W2 + (size_t)(col0 + n) * DIM;

  v8f acc = {};
  #pragma unroll
  for (int k = 0; k < DIM; k += 4) {
    v2f a, b;
    a.x = Ah[n][k + kh];
    a.y = Ah[n][k + kh + 1];
    b.x = Wrow[k + kh];
    b.y = Wrow[k + kh + 1];
    acc = __builtin_amdgcn_wmma_f32_16x16x4_f32(false, a, false, b,
                                                (short)0, acc, false, false);
  }

  int half = lane >> 4;
  float bias = b2[col0 + n];
  #pragma unroll
  for (int r = 0; r < 8; ++r) {
    int row = row0 + r + 8 * half;
    size_t idx = (size_t)row * DIM + col0 + n;
    out[idx] = x[idx] + fmaxf(acc[r] + bias, 0.0f);
  }
}

// ---------------------------------------------------------------------------
extern "C" void kernel_launch(void* const* d_in, const int* in_sizes, int n_in,
                              void* d_out, int out_size, void* d_ws, size_t ws_size,
                              hipStream_t stream) {
  (void)in_sizes; (void)n_in; (void)out_size; (void)ws_size;
  const float*     x     = (const float*)d_in[0];
  const float*     ea    = (const float*)d_in[1];
  const float*     W1    = (const float*)d_in[2];
  const float*     b1    = (const float*)d_in[3];
  const float*     gamma = (const float*)d_in[4];
  const float*     beta  = (const float*)d_in[5];
  const float*     W2    = (const float*)d_in[6];
  const float*     b2    = (const float*)d_in[7];
  const long long* ei    = (const long long*)d_in[8];
  float* out = (float*)d_out;

  float* h    = (float*)d_ws;                       // [N, D]
  float* h1   = h  + (size_t)N_NODES * DIM;         // [N, D]
  float* csum = h1 + (size_t)N_NODES * DIM;         // [D]
  float* csq  = csum + DIM;                         // [D]
  float* sarr = csq  + DIM;                         // [D]
  float* tarr = sarr + DIM;                         // [D]

  // h = x ; stats = 0
  hipMemcpyAsync(h, x, sizeof(float) * (size_t)N_NODES * DIM,
                 hipMemcpyDeviceToDevice, stream);
  hipMemsetAsync(csum, 0, sizeof(float) * 2 * DIM, stream);

  edge_scatter<<<(N_EDGES * 32) / 256, 256, 0, stream>>>(x, ea, ei, h);
  gemm1_wmma<<<N_NODES / 16, 256, 0, stream>>>(h, W1, b1, h1);
  colstats<<<200, 128, 0, stream>>>(h1, csum, csq);
  bnparams<<<1, 128, 0, stream>>>(csum, csq, gamma, beta, sarr, tarr);
  gemm2_wmma<<<N_NODES / 16, 256, 0, stream>>>(h1, sarr, tarr, W2, b2, x, out);
}